// GatingNetwork_23802708754868
// MI455X (gfx1250) — compile-verified
//
#include <hip/hip_runtime.h>
#include <math.h>

// ---------------- problem constants ----------------
#define N_TOK   16384         // B*T
#define C_DIM   2048
#define E_DIM   64
#define KC      32            // K-chunk staged in LDS
#define NCHUNK  (C_DIM / KC)  // 64
#define WAVES   4
#define ROWS_PER_WAVE 16
#define ROWS_PER_WG   (WAVES * ROWS_PER_WAVE)   // 64
#define THREADS (WAVES * 32)                    // 128
// LDS: A[2][64 rows][36] (TDM pads 32->36 dwords/row), B[2][64 e][36 k] (same)
#define ASTR    36
#define ABUF_F  (ROWS_PER_WG * ASTR)            // 2304 floats per A buffer
#define BBUF_F  (E_DIM * ASTR)                  // 2304 floats per B buffer
#define SMEM_F  (2 * ABUF_F + 2 * BBUF_F)       // 9216 floats = 36864 B
#define EPSTR   68                              // epilogue row stride (conflict-free)

typedef float v2f  __attribute__((ext_vector_type(2)));
typedef float v8f  __attribute__((ext_vector_type(8)));
typedef unsigned int u32x4 __attribute__((ext_vector_type(4)));
typedef int   i32x4 __attribute__((ext_vector_type(4)));
typedef int   i32x8 __attribute__((ext_vector_type(8)));

// TDM D# group1 word0: data_size=4B (2<<16) [+ LDS pad cfg]
// padded: pad_enable(1<<20), pad_interval=4 -> every 32 dwords, pad_amount=3 -> 4 dwords
#define PAD_G1W0  ((2 << 16) | (1 << 20) | (4 << 22) | (3 << 25))
#define RAW_G1W0  (2 << 16)

// Issue one 2-D TDM tile load: global[tile1 rows x tile0 elems, row stride0] -> LDS
__device__ __forceinline__ void tdm_load_2d(unsigned ldsAddr, const void* gaddr,
                                            int g1w0, int td0, int td1,
                                            int tile0, int tile1, int stride0)
{
    unsigned long long ga = (unsigned long long)(uintptr_t)gaddr;
    u32x4 g0;
    g0[0] = 1u;                                            // count=1 (valid), user D#
    g0[1] = ldsAddr;                                       // lds_addr [63:32]
    g0[2] = (unsigned)(ga & 0xFFFFFFFFu);                  // global_addr lo
    g0[3] = (unsigned)((ga >> 32) & 0x01FFFFFFu) | (2u << 30); // addr hi | type=2

    i32x8 g1;
    g1[0] = g1w0;                                          // wg_mask=0 | data_size | pad
    g1[1] = (td0 & 0xFFFF) << 16;                          // tensor_dim0[15:0]
    g1[2] = (int)(((unsigned)td0 >> 16) | ((unsigned)(td1 & 0xFFFF) << 16));
    g1[3] = (int)(((unsigned)td1 >> 16) | ((unsigned)tile0 << 16)); // tile_dim0
    g1[4] = tile1 & 0xFFFF;                                // tile_dim1, tile_dim2=0
    g1[5] = stride0;                                       // tensor_dim0_stride lo32
    g1[6] = 0;                                             // stride0 hi | stride1 lo
    g1[7] = 0;                                             // stride1 hi

    i32x4 z4; z4[0] = 0; z4[1] = 0; z4[2] = 0; z4[3] = 0;  // groups 2/3: 2-D tensor
    i32x8 z8;
#pragma unroll
    for (int i = 0; i < 8; ++i) z8[i] = 0;
    __builtin_amdgcn_tensor_load_to_lds(g0, g1, z4, z4, z8, 0);
}

__global__ void gating_init_fb(float* fb) { *fb = 0.0f; }

// Pre-transpose sim [C,E] -> simT [E,C] so TDM can feed B in the same padded
// row-major layout as A (one-time 1 MB of traffic ~ 0.05us; sim is L2-resident).
__global__ void gating_transpose_sim(const float* __restrict__ sim,
                                     float* __restrict__ simT)
{
    const int e = blockIdx.x;
    for (int k = threadIdx.x; k < C_DIM; k += blockDim.x)
        simT[(size_t)e * C_DIM + k] = sim[(size_t)k * E_DIM + e];
}

template <bool BT>
__global__ __launch_bounds__(THREADS)
void gating_wmma_kernel(const float* __restrict__ hs,
                        const float* __restrict__ simB,   // BT ? simT[E,C] : sim[C,E]
                        const float* __restrict__ gates,
                        float* __restrict__ out)
{
    __shared__ float smem[SMEM_F];
    __shared__ float sSig[E_DIM];

    const int tid  = threadIdx.x;
    const int wave = tid >> 5;
    const int lane = tid & 31;
    const int rowBase = blockIdx.x * ROWS_PER_WG;

    if (tid < E_DIM) sSig[tid] = 1.0f / (1.0f + __expf(-gates[tid]));

    const unsigned aLds[2] = { (unsigned)(uintptr_t)&smem[0],
                               (unsigned)(uintptr_t)&smem[ABUF_F] };
    const unsigned bLds[2] = { (unsigned)(uintptr_t)&smem[2 * ABUF_F],
                               (unsigned)(uintptr_t)&smem[2 * ABUF_F + BBUF_F] };

    // ---- TDM: bring chunk 0 into buffer 0 ----
    if (wave == 0) {
        tdm_load_2d(aLds[0], &hs[(size_t)rowBase * C_DIM],
                    PAD_G1W0, C_DIM, N_TOK, KC, ROWS_PER_WG, C_DIM);
        if (BT)
            tdm_load_2d(bLds[0], &simB[0],
                        PAD_G1W0, C_DIM, E_DIM, KC, E_DIM, C_DIM);
        else
            tdm_load_2d(bLds[0], &simB[0],
                        RAW_G1W0, E_DIM, C_DIM, E_DIM, KC, E_DIM);
        __builtin_amdgcn_s_wait_tensorcnt(0);
    }
    __syncthreads();

    // ---- accumulators: 16 rows x 64 experts = 4 tiles of 16x16 f32 ----
    v8f acc[4];
#pragma unroll
    for (int t = 0; t < 4; ++t)
#pragma unroll
        for (int v = 0; v < 8; ++v) acc[t][v] = 0.0f;

    const int li   = lane & 15;
    const int hsel = (lane >> 4) * 2;           // K sub-offset per lane half
    const int aRow = wave * ROWS_PER_WAVE + li;

    for (int c = 0; c < NCHUNK; ++c) {
        const int buf = c & 1;
        // TDM-prefetch chunk c+1 into the other buffer (DMA overlaps compute)
        if (wave == 0 && c + 1 < NCHUNK) {
            const int kbase = (c + 1) * KC;
            tdm_load_2d(aLds[buf ^ 1], &hs[(size_t)rowBase * C_DIM + kbase],
                        PAD_G1W0, C_DIM, N_TOK, KC, ROWS_PER_WG, C_DIM);
            if (BT)
                tdm_load_2d(bLds[buf ^ 1], &simB[kbase],
                            PAD_G1W0, C_DIM, E_DIM, KC, E_DIM, C_DIM);
            else
                tdm_load_2d(bLds[buf ^ 1], &simB[(size_t)kbase * E_DIM],
                            RAW_G1W0, E_DIM, C_DIM, E_DIM, KC, E_DIM);
        }
        // compute chunk c: 8 k-steps x 4 expert tiles of V_WMMA_F32_16X16X4_F32
        {
            const float* As = &smem[buf * ABUF_F];
            const float* Bs = &smem[2 * ABUF_F + buf * BBUF_F];
#pragma unroll
            for (int s = 0; s < KC / 4; ++s) {
                const int k2 = s * 4 + hsel;
                v2f a = *(const v2f*)&As[aRow * ASTR + k2];
#pragma unroll
                for (int t = 0; t < 4; ++t) {
                    v2f b;
                    if (BT) {
                        b = *(const v2f*)&Bs[(t * 16 + li) * ASTR + k2]; // ds_load_b64
                    } else {
                        const float* Bp = &Bs[k2 * E_DIM + t * 16 + li];
                        b.x = Bp[0]; b.y = Bp[E_DIM];
                    }
                    acc[t] = __builtin_amdgcn_wmma_f32_16x16x4_f32(
                        false, a, false, b, (short)0, acc[t], false, false);
                }
            }
        }
        // chunk c+1 must be resident before anyone enters iteration c+1
        if (wave == 0) __builtin_amdgcn_s_wait_tensorcnt(0);
        __syncthreads();
    }

    // ---- epilogue: scatter acc into per-wave LDS tile (reuse smem; final barrier
    //      above guarantees all waves finished reading A/B buffers) ----
#pragma unroll
    for (int t = 0; t < 4; ++t)
#pragma unroll
        for (int v = 0; v < 8; ++v) {
            const int M   = (lane < 16) ? v : v + 8;        // C/D VGPR layout
            const int row = wave * 16 + M;
            const int col = t * 16 + li;
            smem[row * EPSTR + col] = acc[t][v];
        }
    // same-wave LDS ops are in-order; per-wave private region -> no barrier needed

    const float sg0 = sSig[lane];
    const float sg1 = sSig[lane + 32];
    float* probs   = out;
    float* fb      = out + (size_t)N_TOK * E_DIM;
    float* logitsO = fb + 1;
    float* maskO   = logitsO + (size_t)N_TOK * E_DIM;
    const float NEG_MAX = -3.402823466e+38f;

#pragma unroll 1
    for (int r = 0; r < ROWS_PER_WAVE; ++r) {
        const int lrow = wave * 16 + r;
        const size_t grow = (size_t)(rowBase + lrow);
        float aff0 = smem[lrow * EPSTR + lane];
        float aff1 = smem[lrow * EPSTR + lane + 32];
        float l0 = aff0 - sg0, l1 = aff1 - sg1;            // logits
        float g0 = fmaxf(l0, 0.0f), g1 = fmaxf(l1, 0.0f);  // relu
        int m0 = (l0 > 0.0f), m1 = (l1 > 0.0f);
        int cnt = m0 + m1;
        for (int w = 16; w; w >>= 1) cnt += __shfl_xor(cnt, w, 32);

        float a0, a1;
        if (cnt == 0) {
            // inactive row (vanishingly rare): top-32 of logits, ties by index
            int rk0 = 0, rk1 = 0;
            for (int j = 0; j < E_DIM; ++j) {
                float lj = smem[lrow * EPSTR + j] - sSig[j];  // LDS broadcast
                rk0 += (lj > l0) || (lj == l0 && j < lane);
                rk1 += (lj > l1) || (lj == l1 && j < lane + 32);
            }
            a0 = (rk0 < E_DIM / 2) ? 1.0f : 0.0f;
            a1 = (rk1 < E_DIM / 2) ? 1.0f : 0.0f;
            if (lane == 0) atomicAdd(fb, 1.0f);   // integer-valued f32 sum: deterministic
        } else {
            a0 = (float)m0; a1 = (float)m1;
        }

        // masked softmax over gated logits
        float v0 = (a0 > 0.0f) ? g0 : NEG_MAX;
        float v1 = (a1 > 0.0f) ? g1 : NEG_MAX;
        float mx = fmaxf(v0, v1);
        for (int w = 16; w; w >>= 1) mx = fmaxf(mx, __shfl_xor(mx, w, 32));
        float p0 = (a0 > 0.0f) ? __expf(g0 - mx) : 0.0f;
        float p1 = (a1 > 0.0f) ? __expf(g1 - mx) : 0.0f;
        float sum = p0 + p1;
        for (int w = 16; w; w >>= 1) sum += __shfl_xor(sum, w, 32);
        float inv = 1.0f / sum;

        const size_t o = grow * E_DIM;
        probs[o + lane]        = p0 * inv;
        probs[o + lane + 32]   = p1 * inv;
        logitsO[o + lane]      = l0;
        logitsO[o + lane + 32] = l1;
        maskO[o + lane]        = a0;
        maskO[o + lane + 32]   = a1;
    }
}

extern "C" void kernel_launch(void* const* d_in, const int* in_sizes, int n_in,
                              void* d_out, int out_size, void* d_ws, size_t ws_size,
                              hipStream_t stream)
{
    (void)in_sizes; (void)n_in; (void)out_size;
    const float* hs    = (const float*)d_in[0];   // [B,T,C] f32
    const float* sim   = (const float*)d_in[1];   // [C,E]   f32
    const float* gates = (const float*)d_in[2];   // [E]     f32
    float* out = (float*)d_out;                   // probs | fb | logits | mask

    gating_init_fb<<<1, 1, 0, stream>>>(out + (size_t)N_TOK * E_DIM);

    const size_t simT_bytes = (size_t)C_DIM * E_DIM * sizeof(float);
    if (ws_size >= simT_bytes) {
        float* simT = (float*)d_ws;
        gating_transpose_sim<<<E_DIM, 256, 0, stream>>>(sim, simT);
        gating_wmma_kernel<true><<<N_TOK / ROWS_PER_WG, THREADS, 0, stream>>>(
            hs, simT, gates, out);
    } else {
        gating_wmma_kernel<false><<<N_TOK / ROWS_PER_WG, THREADS, 0, stream>>>(
            hs, sim, gates, out);
    }
}